// GAT_36275293782511
// MI455X (gfx1250) — compile-verified
//
#include <hip/hip_runtime.h>
#include <math.h>

#define NEG_SLOPE 0.2f

typedef __attribute__((ext_vector_type(2))) float v2f;
typedef __attribute__((ext_vector_type(8))) float v8f;

// ---------------------------------------------------------------------------
// float atomic max via signed/unsigned integer monotonicity trick
// ---------------------------------------------------------------------------
__device__ __forceinline__ void atomic_max_f32(float* addr, float v) {
  if (v >= 0.f)
    atomicMax((int*)addr, __float_as_int(v));
  else
    atomicMin((unsigned int*)addr, __float_as_uint(v));
}

// ---------------------------------------------------------------------------
// fill
// ---------------------------------------------------------------------------
__global__ void fill_kernel(float* __restrict__ p, long n, float v) {
  long i = (long)blockIdx.x * blockDim.x + threadIdx.x;
  if (i < n) p[i] = v;
}

// ---------------------------------------------------------------------------
// C[M,N] = A[M,K] * B[K,N], f32, via V_WMMA_F32_16X16X4_F32 with LDS staging.
//
// block = 256 threads = 8 wave32 on one WGP.
// Block tile: BM=32 x BN (128 for layer1, 64 for layer2), K-panel KB=16.
// Wave grid: 2 (M) x 4 (N); each wave owns NT 16x16 accumulators.
//
// A tile is staged global->LDS with the gfx1250 async-DMA path
// (global_load_async_to_lds_b128, ASYNCcnt), B tile is transposed through
// VGPRs with global_load_b128. LDS row stride 20 floats keeps float4 LDS
// traffic 16B aligned and float2 fragment reads 8B aligned (ds_load_b64).
//
// Fragment layouts per CDNA5 ISA 7.12.2 (f32 16x16x4):
//   A: lanes 0-15 M=lane K={0,1}; lanes 16-31 M=lane-16 K={2,3}
//   B: lanes 0-15 N=lane K={0,1}; lanes 16-31 N=lane-16 K={2,3}
//   C: v[i]: lanes 0-15 -> (M=i, N=lane); lanes 16-31 -> (M=8+i)
// ---------------------------------------------------------------------------
template <int BN, int NT>
__global__ void gemm_wmma_lds(const float* __restrict__ A,
                              const float* __restrict__ B,
                              float* __restrict__ C,
                              int M, int K, int N) {
  constexpr int BM = 32, KB = 16, STR = 20;  // padded LDS row stride
  __shared__ float As[BM * STR];
  __shared__ float Bst[BN * STR];

  const int t = threadIdx.x;
  const int wave = t >> 5, lane = t & 31;
  const int half = lane >> 4, l15 = lane & 15;
  const int kk = half << 1;                 // K sub-offset 0 or 2
  const int m0 = blockIdx.x * BM;
  const int n0 = blockIdx.y * BN;
  const int wm = (wave & 1) << 4;           // wave M tile (0 or 16)
  const int wn = (wave >> 1) * (NT << 4);   // wave N tile base

  v8f acc[NT];
  const v8f vzero = {0.f, 0.f, 0.f, 0.f, 0.f, 0.f, 0.f, 0.f};
#pragma unroll
  for (int i = 0; i < NT; ++i) acc[i] = vzero;

  for (int k0 = 0; k0 < K; k0 += KB) {
    // ---- stage A tile via async DMA: BM*KB floats, threads 0..127 x 16B ----
    if (t < (BM * KB / 4)) {
      const int r = t >> 2, c4 = (t & 3) << 2;
      const float* gp = &A[(long)(m0 + r) * K + k0 + c4];
      const unsigned lds = (unsigned)(unsigned long)(const void*)&As[r * STR + c4];
      asm volatile("global_load_async_to_lds_b128 %0, %1, off"
                   :
                   : "v"(lds), "v"((unsigned long)gp)
                   : "memory");
      if (k0 + KB < K)  // gfx1250 global_prefetch_b8 for the next K panel
        __builtin_prefetch(gp + KB, 0, 1);
    }
    // ---- cooperative stage B tile transposed: KB x BN floats ----
    constexpr int BV = KB * BN / 4;
#pragma unroll
    for (int g = t; g < BV; g += 256) {
      const int kr = g / (BN / 4);
      const int cg = g - kr * (BN / 4);
      const float* gp = &B[(long)(k0 + kr) * N + n0 + (cg << 2)];
      float4 bv = *(const float4*)gp;
      if (k0 + KB < K) __builtin_prefetch(gp + (long)KB * N, 0, 1);
      Bst[((cg << 2) + 0) * STR + kr] = bv.x;
      Bst[((cg << 2) + 1) * STR + kr] = bv.y;
      Bst[((cg << 2) + 2) * STR + kr] = bv.z;
      Bst[((cg << 2) + 3) * STR + kr] = bv.w;
    }
    // async A copy must land in LDS before anyone reads it
    asm volatile("s_wait_asynccnt 0x0" ::: "memory");
    __syncthreads();

    // ---- WMMA over the K panel ----
#pragma unroll
    for (int kb = 0; kb < KB; kb += 4) {
      v2f a = *(const v2f*)&As[(wm + l15) * STR + kb + kk];
#pragma unroll
      for (int ti = 0; ti < NT; ++ti) {
        v2f b = *(const v2f*)&Bst[(wn + (ti << 4) + l15) * STR + kb + kk];
        acc[ti] = __builtin_amdgcn_wmma_f32_16x16x4_f32(
            false, a, false, b, (short)0, acc[ti], false, false);
      }
    }
    __syncthreads();
  }

  const int rb = half << 3;
#pragma unroll
  for (int ti = 0; ti < NT; ++ti) {
    float* Cp = C + (long)(m0 + wm + rb) * N + n0 + wn + (ti << 4) + l15;
#pragma unroll
    for (int i = 0; i < 8; ++i) Cp[(long)i * N] = acc[ti][i];
  }
}

// ---------------------------------------------------------------------------
// es[n,h] = sum_c h[n,h,c]*a_src[h,c]; ed likewise. One thread per (n,h).
// ---------------------------------------------------------------------------
__global__ void att_coef_kernel(const float* __restrict__ h,
                                const float* __restrict__ a_src,
                                const float* __restrict__ a_dst,
                                float* __restrict__ es,
                                float* __restrict__ ed,
                                int Nn, int H, int C) {
  int idx = blockIdx.x * blockDim.x + threadIdx.x;
  if (idx >= Nn * H) return;
  int n = idx / H, hh = idx - n * H;
  const float* hp = h + (long)n * H * C + hh * C;
  const float* as = a_src + hh * C;
  const float* ad = a_dst + hh * C;
  float s = 0.f, d = 0.f;
  for (int c = 0; c < C; ++c) {
    float v = hp[c];
    s += v * as[c];
    d += v * ad[c];
  }
  es[idx] = s;
  ed[idx] = d;
}

// ---------------------------------------------------------------------------
// Edge passes. ei[0..E) = src, ei[E..2E) = dst; edges E..T are self-loops.
// ---------------------------------------------------------------------------
__device__ __forceinline__ void edge_endpoints(const int* ei, int e, int E,
                                               int& s, int& d) {
  if (e < E) { s = ei[e]; d = ei[E + e]; }
  else       { s = e - E; d = s; }
}

__global__ void edge_max_kernel(const int* __restrict__ ei,
                                const float* __restrict__ es,
                                const float* __restrict__ ed,
                                float* __restrict__ m,
                                int E, int T, int H) {
  int idx = blockIdx.x * blockDim.x + threadIdx.x;
  if (idx >= T * H) return;
  int e = idx / H, hh = idx - e * H;
  int s, d; edge_endpoints(ei, e, E, s, d);
  float v = es[s * H + hh] + ed[d * H + hh];
  v = v > 0.f ? v : NEG_SLOPE * v;
  atomic_max_f32(&m[d * H + hh], v);
}

__global__ void edge_expsum_kernel(const int* __restrict__ ei,
                                   const float* __restrict__ es,
                                   const float* __restrict__ ed,
                                   const float* __restrict__ m,
                                   float* __restrict__ ssum,
                                   int E, int T, int H) {
  int idx = blockIdx.x * blockDim.x + threadIdx.x;
  if (idx >= T * H) return;
  int e = idx / H, hh = idx - e * H;
  int s, d; edge_endpoints(ei, e, E, s, d);
  int dh = d * H + hh;
  float v = es[s * H + hh] + ed[dh];
  v = v > 0.f ? v : NEG_SLOPE * v;
  float mm = m[dh];
  if (!(mm > -INFINITY)) mm = 0.f;  // nodes with no in-edges
  atomicAdd(&ssum[dh], __expf(v - mm));
}

__global__ void edge_scatter_kernel(const int* __restrict__ ei,
                                    const float* __restrict__ es,
                                    const float* __restrict__ ed,
                                    const float* __restrict__ m,
                                    const float* __restrict__ ssum,
                                    const float* __restrict__ hfeat,
                                    float* __restrict__ aggr,
                                    int E, int T, int H, int C) {
  int idx = blockIdx.x * blockDim.x + threadIdx.x;
  if (idx >= T * H) return;
  int e = idx / H, hh = idx - e * H;
  int s, d; edge_endpoints(ei, e, E, s, d);
  int dh = d * H + hh;
  float v = es[s * H + hh] + ed[dh];
  v = v > 0.f ? v : NEG_SLOPE * v;
  float mm = m[dh];
  if (!(mm > -INFINITY)) mm = 0.f;
  float alpha = __expf(v - mm) / (ssum[dh] + 1e-16f);
  const float* hp = hfeat + (long)s * H * C + hh * C;
  float* op = aggr + (long)d * H * C + hh * C;
  for (int c = 0; c < C; ++c)
    atomicAdd(&op[c], hp[c] * alpha);
}

// ---------------------------------------------------------------------------
// out[i] = elu(in[i] + bias[i % F])   /   p[i] += bias[i % F]
// ---------------------------------------------------------------------------
__global__ void elu_bias_kernel(const float* __restrict__ in,
                                const float* __restrict__ bias,
                                float* __restrict__ outp, long n, int F) {
  long i = (long)blockIdx.x * blockDim.x + threadIdx.x;
  if (i >= n) return;
  float v = in[i] + bias[i % F];
  outp[i] = v > 0.f ? v : expm1f(v);
}

__global__ void add_bias_kernel(float* __restrict__ p,
                                const float* __restrict__ bias, long n, int F) {
  long i = (long)blockIdx.x * blockDim.x + threadIdx.x;
  if (i >= n) return;
  p[i] += bias[i % F];
}

// ---------------------------------------------------------------------------
// launch
// ---------------------------------------------------------------------------
extern "C" void kernel_launch(void* const* d_in, const int* in_sizes, int n_in,
                              void* d_out, int out_size, void* d_ws, size_t ws_size,
                              hipStream_t stream) {
  const float* x   = (const float*)d_in[0];
  const int*   ei  = (const int*)  d_in[1];
  const float* W1  = (const float*)d_in[2];
  const float* as1 = (const float*)d_in[3];
  const float* ad1 = (const float*)d_in[4];
  const float* b1  = (const float*)d_in[5];
  const float* W2  = (const float*)d_in[6];
  const float* as2 = (const float*)d_in[7];
  const float* ad2 = (const float*)d_in[8];
  const float* b2  = (const float*)d_in[9];
  float* out = (float*)d_out;

  const int Nn = 20000;           // nodes
  const int E  = in_sizes[1] / 2; // 160000
  const int T  = E + Nn;          // + self loops
  const int F0 = 128, H1 = 32, C1 = 32, F1 = H1 * C1, F2 = 64;

  float* ws = (float*)d_ws;
  float* h1    = ws; ws += (long)Nn * F1;   // xW1 -> later elu output
  float* aggr1 = ws; ws += (long)Nn * F1;
  float* es1   = ws; ws += (long)Nn * H1;
  float* ed1   = ws; ws += (long)Nn * H1;
  float* m1    = ws; ws += (long)Nn * H1;
  float* s1    = ws; ws += (long)Nn * H1;
  float* t2    = ws; ws += (long)Nn * F2;
  float* es2   = ws; ws += Nn;
  float* ed2   = ws; ws += Nn;
  float* m2    = ws; ws += Nn;
  float* s2    = ws; ws += Nn;

  const int BT = 256;
  const float NEG_INF = -INFINITY;
  auto blocks = [](long n, int bt) { return (unsigned)((n + bt - 1) / bt); };

  // ---- init ----
  fill_kernel<<<blocks((long)Nn * F1, BT), BT, 0, stream>>>(aggr1, (long)Nn * F1, 0.f);
  fill_kernel<<<blocks((long)Nn * H1, BT), BT, 0, stream>>>(m1, (long)Nn * H1, NEG_INF);
  fill_kernel<<<blocks((long)Nn * H1, BT), BT, 0, stream>>>(s1, (long)Nn * H1, 0.f);
  fill_kernel<<<blocks((long)Nn * F2, BT), BT, 0, stream>>>(out, (long)Nn * F2, 0.f);
  fill_kernel<<<blocks(Nn, BT), BT, 0, stream>>>(m2, Nn, NEG_INF);
  fill_kernel<<<blocks(Nn, BT), BT, 0, stream>>>(s2, Nn, 0.f);

  // ---- layer 1 ----
  gemm_wmma_lds<128, 2><<<dim3(Nn / 32, F1 / 128), 256, 0, stream>>>(x, W1, h1, Nn, F0, F1);
  att_coef_kernel<<<blocks((long)Nn * H1, BT), BT, 0, stream>>>(h1, as1, ad1, es1, ed1, Nn, H1, C1);
  edge_max_kernel<<<blocks((long)T * H1, BT), BT, 0, stream>>>(ei, es1, ed1, m1, E, T, H1);
  edge_expsum_kernel<<<blocks((long)T * H1, BT), BT, 0, stream>>>(ei, es1, ed1, m1, s1, E, T, H1);
  edge_scatter_kernel<<<blocks((long)T * H1, BT), BT, 0, stream>>>(ei, es1, ed1, m1, s1, h1, aggr1,
                                                                   E, T, H1, C1);
  // h1 <- elu(aggr1 + b1)
  elu_bias_kernel<<<blocks((long)Nn * F1, BT), BT, 0, stream>>>(aggr1, b1, h1, (long)Nn * F1, F1);

  // ---- layer 2 ----
  gemm_wmma_lds<64, 1><<<dim3(Nn / 32, F2 / 64), 256, 0, stream>>>(h1, W2, t2, Nn, F1, F2);
  att_coef_kernel<<<blocks(Nn, BT), BT, 0, stream>>>(t2, as2, ad2, es2, ed2, Nn, 1, F2);
  edge_max_kernel<<<blocks(T, BT), BT, 0, stream>>>(ei, es2, ed2, m2, E, T, 1);
  edge_expsum_kernel<<<blocks(T, BT), BT, 0, stream>>>(ei, es2, ed2, m2, s2, E, T, 1);
  edge_scatter_kernel<<<blocks(T, BT), BT, 0, stream>>>(ei, es2, ed2, m2, s2, t2, out, E, T, 1, F2);
  add_bias_kernel<<<blocks((long)Nn * F2, BT), BT, 0, stream>>>(out, b2, (long)Nn * F2, F2);
}